// RoleDecoder_5025111736730
// MI455X (gfx1250) — compile-verified
//
#include <hip/hip_runtime.h>
#include <math.h>

// ---------------- problem constants ----------------
#define R_   8
#define B_   8
#define S_   512
#define H_   768
#define E_   64
#define NTOK (B_ * S_)      // 4096 tokens

typedef float v2f __attribute__((ext_vector_type(2)));
typedef float v8f __attribute__((ext_vector_type(8)));
typedef unsigned int u32x4 __attribute__((ext_vector_type(4)));
typedef int i32x4 __attribute__((ext_vector_type(4)));
typedef int i32x8 __attribute__((ext_vector_type(8)));

// ---------------- helpers ----------------
__device__ __forceinline__ float waveRedSum(float v) {
#pragma unroll
  for (int o = 16; o > 0; o >>= 1) v += __shfl_down(v, o, 32);
  return v;
}

__device__ __forceinline__ float sigmoidf_(float x) {
  return 1.0f / (1.0f + expf(-x));
}

__device__ __forceinline__ int rfl_(unsigned v) {
  return __builtin_amdgcn_readfirstlane((int)v);
}

// ---------------- TDM: async-load one B chunk (KC rows x 64 cols f32) ----------------
// 2-D tile, row-major in memory (row stride H_ dwords), written to LDS with a
// 1-dword pad after every 64 dwords -> LDS row stride 65 dwords.
#define KC    96
#define LDROW 65                    // 64 data dwords + 1 TDM pad dword
#define CHUNK_DW (KC * LDROW)       // 6240 dwords per buffer

__device__ __forceinline__ void tdm_load_chunk(const float* gsrc, unsigned ldsAddr) {
  unsigned long long ga = (unsigned long long)(size_t)gsrc;
  // --- D# group 0 (128b): count=1 | lds_addr | global_addr[56:0] | type=2 ---
  u32x4 g0;
  g0.x = (unsigned)rfl_(1u);                                    // count=1, user mode
  g0.y = (unsigned)rfl_(ldsAddr);                               // lds_addr (bytes)
  g0.z = (unsigned)rfl_((unsigned)(ga & 0xFFFFFFFFull));        // global_addr lo
  g0.w = (unsigned)rfl_(((unsigned)(ga >> 32) & 0x01FFFFFFu) | (2u << 30)); // hi | type=2
  // --- D# group 1 (256b) ---
  i32x8 g1;
  g1[0] = rfl_((2u << 16)          // data_size = 4 bytes
             | (1u << 20)          // pad_enable
             | (5u << 22));        // pad_interval = 64 dwords (pad_amount=0 -> 1 dword)
  g1[1] = rfl_(64u << 16);         // tensor_dim0 lo16 = 64 (abar addr = 0)
  g1[2] = rfl_((unsigned)KC << 16);// tensor_dim0 hi=0, tensor_dim1 lo16 = KC
  g1[3] = rfl_(64u << 16);         // tensor_dim1 hi=0, tile_dim0 = 64
  g1[4] = rfl_((unsigned)KC);      // tile_dim1 = KC, tile_dim2 = 0
  g1[5] = rfl_((unsigned)H_);      // tensor_dim0_stride lo32 = 768
  g1[6] = 0;
  g1[7] = 0;
  i32x4 gz = {0, 0, 0, 0};
#if __clang_major__ >= 23
  i32x8 gz8 = {0, 0, 0, 0, 0, 0, 0, 0};
  __builtin_amdgcn_tensor_load_to_lds(g0, g1, gz, gz, gz8, 0);
#else
  __builtin_amdgcn_tensor_load_to_lds(g0, g1, gz, gz, 0);
#endif
}

// ---------------- WMMA f32 GEMM ----------------
// D[n,h] = A[n,:] @ Bm[:,h]   (N=4096, K=768, 64-col panels)
// mode 0: D = acc                       (precompute T = tok @ W_top)
// mode 1: D = acc + merged[n]*T[n,h] + bias[h]   (pre_next)
__global__ __launch_bounds__(256)
void gemm_wmma_kernel(const float* __restrict__ A,      // [NTOK, H_]
                      const float* __restrict__ Bm,     // [H_, H_] row-major k*H_+n
                      const float* __restrict__ T,      // [NTOK, H_] (mode 1)
                      const float* __restrict__ merged, // [NTOK]     (mode 1)
                      const float* __restrict__ bias,   // [H_]       (mode 1)
                      float* __restrict__ D,            // [NTOK, H_]
                      int mode) {
  __shared__ float ldsB[2 * CHUNK_DW];   // double-buffered B chunk, row stride LDROW

  const int tid  = threadIdx.x;
  const int wave = tid >> 5;
  const int lane = tid & 31;
  const int nl   = lane & 15;        // N index within 16-wide tile / M row for A
  const int hi   = lane >> 4;        // lane-half selects K pair
  const int colBase = blockIdx.x * 64;
  const int rowBase = blockIdx.y * 128 + wave * 16;

  v8f acc[4] = {};                   // four 16x16 f32 accumulators (16 rows x 64 cols)

  // A operand: row = rowBase + nl, K = kc + kk + 2*hi + {0,1}  (16x4 f32 layout)
  const float* aRow = A + (size_t)(rowBase + nl) * H_ + 2 * hi;
  const unsigned ldsBase = (unsigned)(size_t)(void*)&ldsB[0];
  const int NCHUNK = H_ / KC;        // 8

  if (wave == 0) {
    tdm_load_chunk(Bm + colBase, ldsBase);            // chunk 0 -> buf 0
  }

  for (int c = 0; c < NCHUNK; ++c) {
    if (wave == 0) {
      if (c + 1 < NCHUNK) {
        // kick DMA of next chunk into the other buffer (consumed last iter)
        tdm_load_chunk(Bm + (size_t)(c + 1) * KC * H_ + colBase,
                       ldsBase + (unsigned)(((c + 1) & 1) * CHUNK_DW * 4));
        __builtin_amdgcn_s_wait_tensorcnt(1);         // chunk c has landed
      } else {
        __builtin_amdgcn_s_wait_tensorcnt(0);         // last chunk landed
      }
    }
    __syncthreads();                                  // publish buf[c&1]

    const float* buf = ldsB + (c & 1) * CHUNK_DW;
    const float* a0  = aRow + c * KC;
    if (c + 1 < NCHUNK) __builtin_prefetch(a0 + KC, 0, 3);  // global_prefetch next A

#pragma unroll 2
    for (int kk = 0; kk < KC; kk += 4) {
      v2f a = *(const v2f*)(a0 + kk);
      const int kb = (kk + 2 * hi) * LDROW;
#pragma unroll
      for (int t = 0; t < 4; ++t) {
        const int cidx = t * 16 + nl;
        v2f b;
        b.x = buf[kb + cidx];             // B[K=kk+2hi  ][N]
        b.y = buf[kb + LDROW + cidx];     // B[K=kk+2hi+1][N]
        acc[t] = __builtin_amdgcn_wmma_f32_16x16x4_f32(
            /*neg_a=*/false, a, /*neg_b=*/false, b,
            /*c_mod=*/(short)0, acc[t], /*reuse_a=*/false, /*reuse_b=*/false);
      }
    }
    __syncthreads();                                  // done consuming buf[c&1]
  }

  // C/D layout: VGPR j -> row = rowBase + 8*hi + j, col = colBase + t*16 + nl
  const int row0 = rowBase + 8 * hi;
  if (mode == 0) {
#pragma unroll
    for (int t = 0; t < 4; ++t) {
      const int col = colBase + t * 16 + nl;
#pragma unroll
      for (int j = 0; j < 8; ++j) {
        D[(size_t)(row0 + j) * H_ + col] = acc[t][j];
      }
    }
  } else {
#pragma unroll
    for (int t = 0; t < 4; ++t) {
      const int col = colBase + t * 16 + nl;
      const float bv = bias[col];
#pragma unroll
      for (int j = 0; j < 8; ++j) {
        const int row = row0 + j;
        D[(size_t)row * H_ + col] =
            acc[t][j] + merged[row] * T[(size_t)row * H_ + col] + bv;
      }
    }
  }
}

// ---------------- round-invariant dot products ----------------
// tokDot[n] = dot(tok[n,:], W_single[H:2H]);  entDot[n] = dot(ent[n,:], W_multi[H:2H])
__global__ __launch_bounds__(256)
void fixed_dots_kernel(const float* __restrict__ tok, const float* __restrict__ ent,
                       const float* __restrict__ Ws, const float* __restrict__ Wm,
                       float* __restrict__ tokDot, float* __restrict__ entDot) {
  const int wave = threadIdx.x >> 5, lane = threadIdx.x & 31;
  const int n = blockIdx.x * 8 + wave;
  const float* tp = tok + (size_t)n * H_;
  const float* ep = ent + (size_t)n * H_;
  float d0 = 0.f, d1 = 0.f;
  for (int h = lane; h < H_; h += 32) {
    d0 += tp[h] * Ws[H_ + h];
    d1 += ep[h] * Wm[H_ + h];
  }
  d0 = waveRedSum(d0);
  d1 = waveRedSum(d1);
  if (lane == 0) { tokDot[n] = d0; entDot[n] = d1; }
}

// ---------------- per-round logits + segment-max scatter ----------------
__global__ __launch_bounds__(256)
void round_dots_kernel(const float* __restrict__ sr,      // [NTOK,H_] (offset by r)
                       const float* __restrict__ preCur,  // [NTOK,H_]
                       const float* __restrict__ tokDot, const float* __restrict__ entDot,
                       const float* __restrict__ Ws, const float* __restrict__ Wm,
                       const float* __restrict__ bS, const float* __restrict__ bM,
                       const int* __restrict__ e2t,       // [NTOK]
                       float* __restrict__ singleOut,     // [NTOK]
                       unsigned int* __restrict__ scoreBits) { // [B_*E_]
  const int wave = threadIdx.x >> 5, lane = threadIdx.x & 31;
  const int n = blockIdx.x * 8 + wave;
  const float* sp = sr + (size_t)n * H_;
  const float* pp = preCur + (size_t)n * H_;
  float d0 = 0.f, d1 = 0.f, d2 = 0.f, d3 = 0.f;
  for (int h = lane; h < H_; h += 32) {
    float sv = sp[h], pv = pp[h];
    d0 += sv * Ws[h];
    d1 += sv * Wm[h];
    d2 += pv * Ws[2 * H_ + h];
    d3 += pv * Wm[2 * H_ + h];
  }
  d0 = waveRedSum(d0); d1 = waveRedSum(d1);
  d2 = waveRedSum(d2); d3 = waveRedSum(d3);
  if (lane == 0) {
    float single = sigmoidf_(d0 + tokDot[n] + d2 + bS[0]);
    float multi  = sigmoidf_(d1 + entDot[n] + d3 + bM[0]);
    singleOut[n] = single;
    const int b = n / S_;
    // sigmoid > 0, scores init 0 -> uint-bit atomicMax == segment_max then clamp at 0
    atomicMax(&scoreBits[b * E_ + e2t[n]], __float_as_uint(multi));
  }
}

// ---------------- merge + logits + BCE partial sum ----------------
__global__ __launch_bounds__(256)
void merge_bce_kernel(const float* __restrict__ singleIn,
                      const unsigned int* __restrict__ scoreBits,
                      const int* __restrict__ t2e,
                      const float* __restrict__ labels,   // [NTOK] (offset by r)
                      float* __restrict__ mergedOut,      // [NTOK]
                      float* __restrict__ logitsOut,      // [NTOK] (offset by r)
                      float* __restrict__ lossAcc) {
  __shared__ float red[256];
  const int n = blockIdx.x * 256 + threadIdx.x;
  const int b = n / S_;
  const float pred = __uint_as_float(scoreBits[b * E_ + t2e[n]]);
  const float merged = fmaxf(singleIn[n], pred);
  mergedOut[n] = merged;
  logitsOut[n] = merged;
  const float eps = 1e-7f;
  const float p = fminf(fmaxf(merged, eps), 1.0f - eps);
  const float lab = labels[n];
  const float term = -(lab * logf(p) + (1.0f - lab) * log1pf(-p));
  red[threadIdx.x] = term;
  __syncthreads();
  for (int s = 128; s > 0; s >>= 1) {
    if (threadIdx.x < s) red[threadIdx.x] += red[threadIdx.x + s];
    __syncthreads();
  }
  if (threadIdx.x == 0) atomicAdd(lossAcc, red[0]);
}

// ---------------- small utility kernels ----------------
__global__ void fill_zero_kernel(float* __restrict__ p, int n) {
  int i = blockIdx.x * 256 + threadIdx.x;
  if (i < n) p[i] = 0.0f;
}

__global__ void zero_scores_loss_kernel(unsigned int* __restrict__ scores,
                                        float* __restrict__ lossAcc, int zeroLoss) {
  int i = threadIdx.x;
  if (i < B_ * E_) scores[i] = 0u;
  if (zeroLoss && i == 0) lossAcc[0] = 0.0f;
}

__global__ void mask_sum_kernel(const float* __restrict__ mask, float* __restrict__ out) {
  __shared__ float red[256];
  float v = 0.f;
  for (int i = threadIdx.x; i < NTOK; i += 256) v += mask[i];
  red[threadIdx.x] = v;
  __syncthreads();
  for (int s = 128; s > 0; s >>= 1) {
    if (threadIdx.x < s) red[threadIdx.x] += red[threadIdx.x + s];
    __syncthreads();
  }
  if (threadIdx.x == 0) out[0] = red[0];
}

__global__ void finalize_kernel(const float* __restrict__ lossAcc,
                                const float* __restrict__ maskSum,
                                float* __restrict__ out) {
  if (threadIdx.x == 0) out[0] = lossAcc[0] * maskSum[0] / (float)NTOK;
}

// ---------------- launch ----------------
extern "C" void kernel_launch(void* const* d_in, const int* in_sizes, int n_in,
                              void* d_out, int out_size, void* d_ws, size_t ws_size,
                              hipStream_t stream) {
  const float* role_labels = (const float*)d_in[0];   // [R,B,S]
  const float* summar      = (const float*)d_in[1];   // [R,B,S,H]
  const float* tok         = (const float*)d_in[2];   // [B,S,H]
  const float* ent         = (const float*)d_in[3];   // [B,S,H]
  const float* token_mask  = (const float*)d_in[4];   // [B,S]
  const int*   e2t         = (const int*)d_in[5];     // [B,S]
  const int*   t2e         = (const int*)d_in[6];     // [B,S]
  const float* W_single    = (const float*)d_in[7];   // [3H]
  const float* b_single    = (const float*)d_in[8];
  const float* W_multi     = (const float*)d_in[9];
  const float* b_multi     = (const float*)d_in[10];
  const float* W_ans       = (const float*)d_in[11];  // [2H,H]
  const float* b_ans       = (const float*)d_in[12];  // [H]
  float* out = (float*)d_out;                         // [1 + R*B*S]

  float* ws = (float*)d_ws;
  size_t off = 0;
  float* T      = ws + off; off += (size_t)NTOK * H_;   // tok @ W_top (round invariant)
  float* pre0   = ws + off; off += (size_t)NTOK * H_;
  float* pre1   = ws + off; off += (size_t)NTOK * H_;
  float* tokDot = ws + off; off += NTOK;
  float* entDot = ws + off; off += NTOK;
  float* single = ws + off; off += NTOK;
  float* merged = ws + off; off += NTOK;
  unsigned int* scores = (unsigned int*)(ws + off); off += B_ * E_;
  float* lossAcc = ws + off; off += 1;
  float* maskSum = ws + off; off += 1;

  // init: pre_answer = 0, loss = 0, mask_sum
  fill_zero_kernel<<<(NTOK * H_ + 255) / 256, 256, 0, stream>>>(pre0, NTOK * H_);
  mask_sum_kernel<<<1, 256, 0, stream>>>(token_mask, maskSum);
  fixed_dots_kernel<<<NTOK / 8, 256, 0, stream>>>(tok, ent, W_single, W_multi,
                                                  tokDot, entDot);
  dim3 gg(H_ / 64, NTOK / 128);
  // T = tok @ W_ans[0:H,:]
  gemm_wmma_kernel<<<gg, 256, 0, stream>>>(tok, W_ans, nullptr, nullptr, nullptr, T, 0);

  float* preCur = pre0;
  float* preNext = pre1;
  for (int r = 0; r < R_; ++r) {
    zero_scores_loss_kernel<<<1, 512, 0, stream>>>(scores, lossAcc, r == 0 ? 1 : 0);
    round_dots_kernel<<<NTOK / 8, 256, 0, stream>>>(
        summar + (size_t)r * NTOK * H_, preCur, tokDot, entDot,
        W_single, W_multi, b_single, b_multi, e2t, single, scores);
    merge_bce_kernel<<<NTOK / 256, 256, 0, stream>>>(
        single, scores, t2e, role_labels + (size_t)r * NTOK,
        merged, out + 1 + (size_t)r * NTOK, lossAcc);
    if (r + 1 < R_) {
      // pre_next = merged*T + preCur @ W_ans[H:2H,:] + b_ans
      gemm_wmma_kernel<<<gg, 256, 0, stream>>>(preCur, W_ans + (size_t)H_ * H_,
                                               T, merged, b_ans, preNext, 1);
      float* tmp = preCur; preCur = preNext; preNext = tmp;
    }
  }
  finalize_kernel<<<1, 64, 0, stream>>>(lossAcc, maskSum, out);
}